// GCN_30691836297408
// MI455X (gfx1250) — compile-verified
//
#include <hip/hip_runtime.h>
#include <math.h>

#define D 128
#define GRAPHS 256
#define SCAN_B 256

typedef float v2f __attribute__((ext_vector_type(2)));
typedef float v8f __attribute__((ext_vector_type(8)));

// ---------------------------------------------------------------------------
// zero-fill (bit pattern 0 works for both f32 and i32 scratch)
// ---------------------------------------------------------------------------
__global__ void __launch_bounds__(256) zero_b32(unsigned* __restrict__ p, long long n) {
    long long i = (long long)blockIdx.x * blockDim.x + threadIdx.x;
    long long stride = (long long)gridDim.x * blockDim.x;
    for (; i < n; i += stride) p[i] = 0u;
}

// ---------------------------------------------------------------------------
// CSR build (by destination), reused for both conv layers
// ---------------------------------------------------------------------------
__global__ void __launch_bounds__(256)
edge_hist(const int* __restrict__ dst, int* __restrict__ deg, int E) {
    int i = blockIdx.x * blockDim.x + threadIdx.x;
    int stride = gridDim.x * blockDim.x;
    for (int e = i; e < E; e += stride) atomicAdd(&deg[dst[e]], 1);
}

// per-block inclusive scan -> exclusive-within-block + block sums
__global__ void __launch_bounds__(SCAN_B)
scan_block(const int* __restrict__ deg, int* __restrict__ excl,
           int* __restrict__ bsum, int n) {
    __shared__ int s[SCAN_B];
    const int tid = threadIdx.x;
    const int i = blockIdx.x * SCAN_B + tid;
    const int v = (i < n) ? deg[i] : 0;
    s[tid] = v;
    __syncthreads();
#pragma unroll
    for (int off = 1; off < SCAN_B; off <<= 1) {
        int t = (tid >= off) ? s[tid - off] : 0;
        __syncthreads();
        s[tid] += t;
        __syncthreads();
    }
    if (i < n) excl[i] = s[tid] - v;
    if (tid == SCAN_B - 1) bsum[blockIdx.x] = s[tid];
}

// single-block exclusive scan of block sums (nb <= 512; N<=131072 here)
__global__ void __launch_bounds__(512) scan_top(int* __restrict__ bsum, int nb) {
    __shared__ int s[512];
    const int tid = threadIdx.x;
    const int v = (tid < nb) ? bsum[tid] : 0;
    s[tid] = v;
    __syncthreads();
#pragma unroll
    for (int off = 1; off < 512; off <<= 1) {
        int t = (tid >= off) ? s[tid - off] : 0;
        __syncthreads();
        s[tid] += t;
        __syncthreads();
    }
    if (tid < nb) bsum[tid] = s[tid] - v;   // exclusive, in place
}

__global__ void __launch_bounds__(256)
scan_add(const int* __restrict__ excl, const int* __restrict__ bsum,
         int* __restrict__ rowptr, int* __restrict__ cursor, int n, int E) {
    const int i = blockIdx.x * blockDim.x + threadIdx.x;
    if (i < n) {
        int r = excl[i] + bsum[i >> 8];
        rowptr[i] = r;
        cursor[i] = r;
    }
    if (i == n) rowptr[n] = E;
}

__global__ void __launch_bounds__(256)
edge_fill(const int* __restrict__ src, const int* __restrict__ dst,
          int* __restrict__ cursor, int* __restrict__ colidx, int E) {
    int i = blockIdx.x * blockDim.x + threadIdx.x;
    int stride = gridDim.x * blockDim.x;
    for (int e = i; e < E; e += stride) {
        int p = atomicAdd(&cursor[dst[e]], 1);
        colidx[p] = src[e];
    }
}

// ---------------------------------------------------------------------------
// H[row, :] = act(X[row, :]) @ W  (W is 128x128), fp32 WMMA 16x16x4.
// 256-thread block = 8 waves; wave owns a 16-row strip across all 8 N-tiles.
// W staged in LDS in K-pair-interleaved layout: pair j holds (W[2j][c], W[2j+1][c])
// contiguous -> each B fragment is one aligned ds_load_b64, no repacking.
// ---------------------------------------------------------------------------
template <bool RELU_IN>
__global__ void __launch_bounds__(256)
gemm_nodes_128(const float* __restrict__ X, const float* __restrict__ W,
               float* __restrict__ H, int nrows) {
    __shared__ float Ws[D * D];   // 65536 bytes, pair-interleaved

    const int t = threadIdx.x;
    {
        const float4* Wv = (const float4*)W;
#pragma unroll
        for (int i = 0; i < 16; ++i) {
            float4 v = Wv[i * 256 + t];
            int idx = (i * 256 + t) * 4;          // element index, row-major
            int k = idx >> 7, c = idx & 127;      // all 4 elems share k
            int j = k >> 1, odd = k & 1;
            Ws[(((j << 7) + c + 0) << 1) | odd] = v.x;
            Ws[(((j << 7) + c + 1) << 1) | odd] = v.y;
            Ws[(((j << 7) + c + 2) << 1) | odd] = v.z;
            Ws[(((j << 7) + c + 3) << 1) | odd] = v.w;
        }
    }
    __syncthreads();

    const int wave = t >> 5;
    const int lane = t & 31;
    const int row0 = blockIdx.x * 128 + wave * 16;
    if (row0 >= nrows) return;              // whole wave exits together

    const int m     = lane & 15;
    const int half  = lane >> 4;            // 0: K=0,1  1: K=2,3
    const int khalf = half << 1;

    v8f acc[8] = {};
    const float* xrow = X + (size_t)(row0 + m) * D;
    const v2f* Bp = (const v2f*)Ws;

#pragma unroll 4
    for (int k0 = 0; k0 < D; k0 += 4) {
        const int kk = k0 + khalf;
        v2f a = *(const v2f*)(xrow + kk);
        if (RELU_IN) { a.x = fmaxf(a.x, 0.0f); a.y = fmaxf(a.y, 0.0f); }
        const int j = (k0 >> 1) + half;     // K-pair index for this half-wave
#pragma unroll
        for (int n = 0; n < 8; ++n) {
            v2f b = Bp[(j << 7) + n * 16 + m];   // one ds_load_b64
            acc[n] = __builtin_amdgcn_wmma_f32_16x16x4_f32(
                false, a, false, b, (short)0, acc[n], false, false);
        }
    }

    const int rbase = half * 8;
#pragma unroll
    for (int n = 0; n < 8; ++n) {
#pragma unroll
        for (int r = 0; r < 8; ++r) {
            H[(size_t)(row0 + rbase + r) * D + n * 16 + m] = acc[n][r];
        }
    }
}

// ---------------------------------------------------------------------------
// CSR gather-sum: agg[n,:] = sum_{e in [rowptr[n],rowptr[n+1])} h[colidx[e],:]
// one wave per node, lane holds a float4 (32*4 = 128 dims); no atomics, and
// every node row is written so no pre-zeroing of agg is needed.
// ---------------------------------------------------------------------------
__global__ void __launch_bounds__(256)
aggregate_csr(const float* __restrict__ h, const int* __restrict__ rowptr,
              const int* __restrict__ colidx, float* __restrict__ agg, int N) {
    const int gtid   = blockIdx.x * blockDim.x + threadIdx.x;
    const int wave   = gtid >> 5;
    const int lane   = threadIdx.x & 31;
    const int nwaves = (gridDim.x * blockDim.x) >> 5;
    for (int n = wave; n < N; n += nwaves) {
        const int beg = rowptr[n];
        const int end = rowptr[n + 1];
        float4 acc = make_float4(0.f, 0.f, 0.f, 0.f);
        int e = beg;
        for (; e + 1 < end; e += 2) {
            const int s0 = colidx[e];
            const int s1 = colidx[e + 1];
            float4 a = ((const float4*)(h + (size_t)s0 * D))[lane];
            float4 b = ((const float4*)(h + (size_t)s1 * D))[lane];
            acc.x += a.x + b.x; acc.y += a.y + b.y;
            acc.z += a.z + b.z; acc.w += a.w + b.w;
        }
        if (e < end) {
            const int s0 = colidx[e];
            float4 a = ((const float4*)(h + (size_t)s0 * D))[lane];
            acc.x += a.x; acc.y += a.y; acc.z += a.z; acc.w += a.w;
        }
        ((float4*)(agg + (size_t)n * D))[lane] = acc;
    }
}

// ---------------------------------------------------------------------------
// mean-pool + readout, atomic-free: batch is sorted, so block g binary-searches
// its node range, reduces over it, then out[g] = sigmoid(mean . Wout + bout).
// ---------------------------------------------------------------------------
__global__ void __launch_bounds__(128)
graph_pool(const float* __restrict__ h, const int* __restrict__ batch,
           const float* __restrict__ Wout, const float* __restrict__ bout,
           float* __restrict__ out, int N) {
    const int g = blockIdx.x;
    const int d = threadIdx.x;          // 0..127, one dim per thread

    // lower_bound(batch, key)
    int start, end;
    {
        int lo = 0, hi = N;
        while (lo < hi) { int mid = (lo + hi) >> 1; if (batch[mid] < g) lo = mid + 1; else hi = mid; }
        start = lo;
        hi = N;
        while (lo < hi) { int mid = (lo + hi) >> 1; if (batch[mid] < g + 1) lo = mid + 1; else hi = mid; }
        end = lo;
    }

    float acc = 0.0f;
    for (int n = start; n < end; ++n) acc += h[(size_t)n * D + d];
    const float cnt = (float)(end - start);
    const float mean = acc / fmaxf(cnt, 1.0f);
    float partial = mean * Wout[d];

#pragma unroll
    for (int off = 16; off > 0; off >>= 1) partial += __shfl_xor(partial, off, 32);

    __shared__ float wsum[4];
    const int lane = d & 31, w = d >> 5;
    if (lane == 0) wsum[w] = partial;
    __syncthreads();
    if (d == 0) {
        float tot = wsum[0] + wsum[1] + wsum[2] + wsum[3];
        out[g] = 1.0f / (1.0f + expf(-(tot + bout[0])));
    }
}

// ---------------------------------------------------------------------------
extern "C" void kernel_launch(void* const* d_in, const int* in_sizes, int n_in,
                              void* d_out, int out_size, void* d_ws, size_t ws_size,
                              hipStream_t stream) {
    const float* x     = (const float*)d_in[0];
    const int*   ei    = (const int*)  d_in[1];   // [2, E] : row0=src, row1=dst
    const int*   batch = (const int*)  d_in[2];
    const float* W1    = (const float*)d_in[3];
    const float* W2    = (const float*)d_in[4];
    const float* Wout  = (const float*)d_in[5];
    const float* bout  = (const float*)d_in[6];
    float*       out   = (float*)d_out;

    const int N = in_sizes[0] / D;
    const int E = in_sizes[1] / 2;
    const int* src = ei;
    const int* dst = ei + E;

    // workspace: h[N*D] | agg[N*D] | deg[N] | excl[N] | rowptr[N+1] | cursor[N] | colidx[E] | bsum[512]
    float* h      = (float*)d_ws;
    float* agg    = h + (size_t)N * D;
    int*   deg    = (int*)(agg + (size_t)N * D);
    int*   excl   = deg + N;
    int*   rowptr = excl + N;
    int*   cursor = rowptr + (N + 1);
    int*   colidx = cursor + N;
    int*   bsum   = colidx + E;

    const dim3 blk256(256), blk128(128), blk512(512);
    const int gemmGrid = (N + 127) / 128;
    const int nb = (N + SCAN_B - 1) / SCAN_B;          // <= 512

    // ---- CSR build (once; reused by both conv layers) ----
    zero_b32<<<256, blk256, 0, stream>>>((unsigned*)deg, N);
    edge_hist<<<1024, blk256, 0, stream>>>(dst, deg, E);
    scan_block<<<nb, blk256, 0, stream>>>(deg, excl, bsum, N);
    scan_top<<<1, blk512, 0, stream>>>(bsum, nb);
    scan_add<<<(N + 256) / 256, blk256, 0, stream>>>(excl, bsum, rowptr, cursor, N, E);
    edge_fill<<<1024, blk256, 0, stream>>>(src, dst, cursor, colidx, E);

    // ---- layer 1: h = x @ W1 ; agg = csr_gather_sum(h) ----
    gemm_nodes_128<false><<<gemmGrid, blk256, 0, stream>>>(x, W1, h, N);
    aggregate_csr<<<2048, blk256, 0, stream>>>(h, rowptr, colidx, agg, N);

    // ---- layer 2: h = relu(agg) @ W2 ; agg = csr_gather_sum(h) ----
    gemm_nodes_128<true><<<gemmGrid, blk256, 0, stream>>>(agg, W2, h, N);
    aggregate_csr<<<2048, blk256, 0, stream>>>(h, rowptr, colidx, agg, N);

    // ---- mean pool + sigmoid readout ----
    graph_pool<<<GRAPHS, blk128, 0, stream>>>(agg, batch, Wout, bout, out, N);
}